// SAGE_89850715833230
// MI455X (gfx1250) — compile-verified
//
#include <hip/hip_runtime.h>

typedef __attribute__((ext_vector_type(2))) float v2f;
typedef __attribute__((ext_vector_type(8))) float v8f;

#define FEAT 64
// Pair-row stride (floats) for the swizzled weight layout in LDS.
// 160 = 128 + 32 ensures pair p and pair p+1 map to opposite 32-bank halves,
// making the per-fragment ds_load_b64 (lanes 0-15 read pair 2k, lanes 16-31
// read pair 2k+1) conflict-free across the 64 LDS banks.
#define PAIR_STRIDE 160
#define W_LDS_SIZE (32 * PAIR_STRIDE)   // 32 K-pairs

// ---------------------------------------------------------------------------
// Degree count: one thread per edge, float atomic into deg[dst]
// ---------------------------------------------------------------------------
__global__ void deg_kernel(const int* __restrict__ dst, float* __restrict__ deg, int nE) {
    int e = blockIdx.x * blockDim.x + threadIdx.x;
    if (e < nE) atomicAdd(&deg[dst[e]], 1.0f);
}

// In-place deg -> inv_deg (zero-in-degree nodes -> 0, per DGL semantics)
__global__ void inv_kernel(float* __restrict__ deg, int n) {
    int i = blockIdx.x * blockDim.x + threadIdx.x;
    if (i < n) {
        float d = deg[i];
        deg[i] = (d > 0.0f) ? (1.0f / d) : 0.0f;
    }
}

// ---------------------------------------------------------------------------
// Edge aggregation: one wave32 per edge; lane moves a float2 (features
// 2*lane, 2*lane+1). Gather is a coalesced 256B row read (L2-resident:
// feature matrix is 25.6MB << 192MB L2); scatter is 2 f32 atomics.
// ---------------------------------------------------------------------------
__global__ void agg_kernel(const float* __restrict__ feat,
                           const int* __restrict__ src,
                           const int* __restrict__ dst,
                           float* __restrict__ agg, int nE) {
    int t    = blockIdx.x * blockDim.x + threadIdx.x;
    int e    = t >> 5;
    int lane = t & 31;
    if (e >= nE) return;
    int s = src[e];
    int d = dst[e];
    const float2* xr = (const float2*)(feat + (size_t)s * FEAT);
    float2 v = xr[lane];
    float* ar = agg + (size_t)d * FEAT + lane * 2;
    atomicAdd(ar,     v.x);
    atomicAdd(ar + 1, v.y);
}

// ---------------------------------------------------------------------------
// Fused SAGE layer GEMM:  Out = act( X@Ws + (Agg*inv_deg)@Wn + b )
// Block = 256 threads = 8 waves. Weights staged in LDS *pre-swizzled in
// fragment order*: K-rows (2p, 2p+1) interleaved per column, so a WMMA B
// fragment {W[kr][col], W[kr+1][col]} is a single aligned ds_load_b64
// (no repack movs), with PAIR_STRIDE chosen for bank-conflict freedom.
//
// Each wave owns one 16-row node block and 4 f32 16x16 accumulators covering
// the 64 output features: 16 K-steps x 4 tiles x 2 matrices = 128 WMMAs.
//
// f32 WMMA fragment layout (ISA 7.12.2):
//   A (16x4): lane&15 selects row M; the two VGPRs hold K = 2*(lane>>4), +1.
//   B (4x16): symmetric, lane&15 selects column N.
//   C/D (16x16): VGPR v, lane l -> M = v + 8*(l>>4), N = l&15.
// ---------------------------------------------------------------------------
__launch_bounds__(256)
__global__ void sage_gemm(const float* __restrict__ X,
                          const float* __restrict__ Agg,
                          const float* __restrict__ inv_deg,
                          const float* __restrict__ Ws,
                          const float* __restrict__ Wn,
                          const float* __restrict__ bias,
                          float* __restrict__ Out,
                          int nNodes, int doRelu) {
    __shared__ float sWs[W_LDS_SIZE];
    __shared__ float sWn[W_LDS_SIZE];
    __shared__ float sB[FEAT];

    int tid = threadIdx.x;
    // Swizzled staging: W[k][c] -> s[(k>>1)*PAIR_STRIDE + c*2 + (k&1)]
    for (int i = tid; i < FEAT * FEAT; i += 256) {
        int k = i >> 6;
        int c = i & 63;
        int d = (k >> 1) * PAIR_STRIDE + c * 2 + (k & 1);
        sWs[d] = Ws[i];
        sWn[d] = Wn[i];
    }
    if (tid < FEAT) sB[tid] = bias[tid];
    __syncthreads();

    int wave = tid >> 5;
    int lane = tid & 31;
    int base = (blockIdx.x * 8 + wave) * 16;   // first node row of this wave's tile
    if (base >= nNodes) return;                // wave-uniform exit (EXEC stays all-1s)

    int m    = lane & 15;    // M (for A) / N (for B) index
    int koff = lane >> 4;    // selects K pair {0,1} vs {2,3} within a 4-wide step
    int row  = base + m;

    float idg = inv_deg[row];
    const float2* xr = (const float2*)(X   + (size_t)row * FEAT);
    const float2* gr = (const float2*)(Agg + (size_t)row * FEAT);

    v8f acc[4] = {v8f{}, v8f{}, v8f{}, v8f{}};

    for (int k = 0; k < 16; ++k) {            // K = 64 in steps of 4
        float2 ax = xr[2 * k + koff];
        float2 ag = gr[2 * k + koff];
        v2f aS; aS[0] = ax.x;        aS[1] = ax.y;
        v2f aN; aN[0] = ag.x * idg;  aN[1] = ag.y * idg;
        int pairBase = (2 * k + koff) * PAIR_STRIDE;   // K-pair row for this half-wave
#pragma unroll
        for (int t = 0; t < 4; ++t) {         // 4 output tiles of 16 cols
            int off = pairBase + (t * 16 + m) * 2;
            v2f bS = *(const v2f*)&sWs[off];   // one aligned ds_load_b64
            v2f bN = *(const v2f*)&sWn[off];
            acc[t] = __builtin_amdgcn_wmma_f32_16x16x4_f32(
                false, aS, false, bS, (short)0, acc[t], false, false);
            acc[t] = __builtin_amdgcn_wmma_f32_16x16x4_f32(
                false, aN, false, bN, (short)0, acc[t], false, false);
        }
    }

    // Store: VGPR v, lane l -> row base + v + 8*(l>>4), col t*16 + (l&15).
#pragma unroll
    for (int t = 0; t < 4; ++t) {
        int c = t * 16 + m;
        float b = sB[c];
#pragma unroll
        for (int v = 0; v < 8; ++v) {
            int r = base + v + 8 * koff;
            float val = acc[t][v] + b;
            if (doRelu) val = fmaxf(val, 0.0f);
            Out[(size_t)r * FEAT + c] = val;
        }
    }
}

// ---------------------------------------------------------------------------
// Host-side orchestration (graph-capture safe: only async ops on `stream`)
// ---------------------------------------------------------------------------
extern "C" void kernel_launch(void* const* d_in, const int* in_sizes, int n_in,
                              void* d_out, int out_size, void* d_ws, size_t ws_size,
                              hipStream_t stream) {
    const float* x   = (const float*)d_in[0];
    const int*   src = (const int*)  d_in[1];
    const int*   dst = (const int*)  d_in[2];
    const float* Ws1 = (const float*)d_in[3];
    const float* Wn1 = (const float*)d_in[4];
    const float* b1  = (const float*)d_in[5];
    const float* Ws2 = (const float*)d_in[6];
    const float* Wn2 = (const float*)d_in[7];
    const float* b2  = (const float*)d_in[8];
    float* out = (float*)d_out;

    const int N = in_sizes[0] / FEAT;
    const int E = in_sizes[1];

    // Workspace layout: [deg/inv_deg: N] [agg: N*64] [h: N*64]  (~51.6 MB)
    float* deg = (float*)d_ws;
    float* agg = deg + N;
    float* h   = agg + (size_t)N * FEAT;

    const int THREADS = 256;
    const int edgeWaveBlocks = (E + 7) / 8;               // 32 lanes per edge
    const int nodeBlocks     = (N + 15) / 16;
    const int gemmBlocks     = (nodeBlocks + 7) / 8;      // 8 waves per block

    // Zero deg + agg (contiguous), count degrees, invert.
    hipMemsetAsync(deg, 0, sizeof(float) * (size_t)N * (1 + FEAT), stream);
    deg_kernel<<<(E + THREADS - 1) / THREADS, THREADS, 0, stream>>>(dst, deg, E);
    inv_kernel<<<(N + THREADS - 1) / THREADS, THREADS, 0, stream>>>(deg, N);

    // Layer 1: aggregate x, then h = relu(x@Ws1 + (agg/deg)@Wn1 + b1)
    agg_kernel<<<edgeWaveBlocks, THREADS, 0, stream>>>(x, src, dst, agg, E);
    sage_gemm<<<gemmBlocks, THREADS, 0, stream>>>(x, agg, deg, Ws1, Wn1, b1, h, N, 1);

    // Layer 2: re-zero agg, aggregate h, then out = h@Ws2 + (agg/deg)@Wn2 + b2
    hipMemsetAsync(agg, 0, sizeof(float) * (size_t)N * FEAT, stream);
    agg_kernel<<<edgeWaveBlocks, THREADS, 0, stream>>>(h, src, dst, agg, E);
    sage_gemm<<<gemmBlocks, THREADS, 0, stream>>>(h, agg, deg, Ws2, Wn2, b2, out, N, 0);
}